// GearNet_7524782702912
// MI455X (gfx1250) — compile-verified
//
#include <hip/hip_runtime.h>
#include <hip/hip_bf16.h>

#define NN 65536   // nodes
#define NE 262144  // edges
#define DD 128     // dim
#define RR 7       // relations
#define GG 32      // graphs
#define RD (RR * DD)  // 896

typedef float v2f __attribute__((ext_vector_type(2)));
typedef float v8f __attribute__((ext_vector_type(8)));

// meta layout (ints): [0..7]=cnt, [8..16]=off, [20..28]=toff(64-edge tiles), [32..39]=cursor
#define META_CNT 0
#define META_OFF 8
#define META_TOFF 20
#define META_CUR 32

__device__ __forceinline__ v8f wmma4(v2f a, v2f b, v8f c) {
  // D = A(16x4 f32) * B(4x16 f32) + C(16x16 f32)
  return __builtin_amdgcn_wmma_f32_16x16x4_f32(false, a, false, b, (short)0, c,
                                               false, false);
}

// Async DMA: global -> LDS, 16 bytes per lane, tracked by ASYNCcnt.
__device__ __forceinline__ void async_b128_to_lds(void* lds_ptr, const void* gptr) {
  const unsigned loff = (unsigned)(size_t)lds_ptr;        // LDS aperture: addr[31:0]
  const unsigned long long ga = (unsigned long long)(size_t)gptr;
  asm volatile("global_load_async_to_lds_b128 %0, %1, off" ::"v"(loff), "v"(ga)
               : "memory");
}
__device__ __forceinline__ void wait_async_all() {
  asm volatile("s_wait_asynccnt 0x0" ::: "memory");
}

// Stage a 128x128 f32 weight tile into LDS in k-pair-interleaved layout:
//   ldsW2[(p*DD + n)*2 + q] = W[(2p+q)*rowStride + n]
// so a WMMA B-frag (k,k+1)@n is one aligned ds_load_b64 into an even VGPR pair.
__device__ __forceinline__ void stage_w_pairs(float* __restrict__ ldsW2,
                                              const float* __restrict__ Wsrc,
                                              int rowStride) {
  for (int idx = threadIdx.x; idx < 64 * 32; idx += 256) {
    const int p = idx >> 5, n4 = (idx & 31) * 4;
    const float4 g0 = *(const float4*)(Wsrc + (size_t)(2 * p) * rowStride + n4);
    const float4 g1 = *(const float4*)(Wsrc + (size_t)(2 * p + 1) * rowStride + n4);
    float* d = ldsW2 + (size_t)(p * DD + n4) * 2;  // 8 consecutive floats
    d[0] = g0.x; d[1] = g1.x;
    d[2] = g0.y; d[3] = g1.y;
    d[4] = g0.z; d[5] = g1.z;
    d[6] = g0.w; d[7] = g1.w;
  }
}

__device__ __forceinline__ v2f bfrag(const float* __restrict__ ldsW2, int p, int n) {
  return *(const v2f*)(ldsW2 + (size_t)(p * DD + n) * 2);  // ds_load_b64
}

__global__ __launch_bounds__(64) void k_meta_zero(int* meta) {
  meta[threadIdx.x] = 0;
}

__global__ __launch_bounds__(256) void k_hist(const int* __restrict__ rel,
                                              int* __restrict__ meta) {
  const int e = blockIdx.x * 256 + threadIdx.x;
  atomicAdd(&meta[META_CNT + rel[e]], 1);
}

__global__ __launch_bounds__(64) void k_scan(int* meta) {
  if (threadIdx.x == 0) {
    int off = 0, toff = 0;
    meta[META_OFF + 0] = 0;
    meta[META_TOFF + 0] = 0;
    for (int r = 0; r < RR; ++r) {
      meta[META_CUR + r] = off;
      const int c = meta[META_CNT + r];
      off += c;
      toff += (c + 63) >> 6;
      meta[META_OFF + r + 1] = off;
      meta[META_TOFF + r + 1] = toff;
    }
  }
}

__global__ __launch_bounds__(256) void k_fill(const int* __restrict__ rel,
                                              int* __restrict__ meta,
                                              int* __restrict__ order) {
  const int e = blockIdx.x * 256 + threadIdx.x;
  const int pos = atomicAdd(&meta[META_CUR + rel[e]], 1);
  order[pos] = e;
}

// H = X @ W_self + b_self.  Workgroup: 8 waves, 128 rows; W in LDS (64KB,
// pair-interleaved).
__global__ __launch_bounds__(256) void k_gemm_self(const float* __restrict__ X,
                                                   const float* __restrict__ W,
                                                   const float* __restrict__ bias,
                                                   float* __restrict__ H) {
  extern __shared__ float lds[];
  stage_w_pairs(lds, W, DD);
  __syncthreads();

  const int wave = threadIdx.x >> 5, lane = threadIdx.x & 31;
  const int row = lane & 15, hi = lane >> 4;
  const int m0 = blockIdx.x * 128 + wave * 16;
  const float* __restrict__ xr = X + (size_t)(m0 + row) * DD;

  v8f acc[8] = {};
#pragma unroll
  for (int kk = 0; kk < 32; ++kk) {
    // A layout: lanes 0-15 hold (K0,K1), lanes 16-31 hold (K2,K3)
    const int kb = kk * 4 + 2 * hi;
    const v2f a = *(const v2f*)(xr + kb);  // global_load_b64
    const int p = 2 * kk + hi;             // k-pair index in interleaved LDS
#pragma unroll
    for (int nt = 0; nt < 8; ++nt)
      acc[nt] = wmma4(a, bfrag(lds, p, nt * 16 + row), acc[nt]);
  }
#pragma unroll
  for (int nt = 0; nt < 8; ++nt) {
    const float bb = bias[nt * 16 + row];
#pragma unroll
    for (int i = 0; i < 8; ++i)  // D VGPR i -> row i (low lanes) / i+8 (high)
      H[(size_t)(m0 + i + 8 * hi) * DD + nt * 16 + row] = acc[nt][i] + bb;
  }
}

// Per-relation edge messages: 64 same-relation edges per workgroup.
// LDS: W_r (16384 f, pair-interleaved) + 64 gathered X rows (8192 f, async DMA)
//      + 64 src + 64 out indices.
__global__ __launch_bounds__(256) void k_msg(
    const float* __restrict__ X, const float* __restrict__ Wrel,
    const float* __restrict__ brel, const int* __restrict__ order,
    const int* __restrict__ nin, const int* __restrict__ nout,
    const int* __restrict__ meta, float* __restrict__ H) {
  extern __shared__ float lds[];
  float* ldsW = lds;               // [64][128][2] pair-interleaved W_r
  float* ldsX = lds + DD * DD;     // [64][128] gathered rows
  int* ldsSrc = (int*)(lds + DD * DD + 64 * DD);
  int* ldsOut = ldsSrc + 64;

  const int wg = blockIdx.x;
  if (wg >= meta[META_TOFF + RR]) return;
  int r = 0;
#pragma unroll
  for (int i = 1; i < RR; ++i)
    if (wg >= meta[META_TOFF + i]) r = i;
  const int ebase = meta[META_OFF + r] + (wg - meta[META_TOFF + r]) * 64;
  const int eend = meta[META_OFF + r + 1];

  if (threadIdx.x < 64) {
    const int e = ebase + threadIdx.x;
    int s = 0, o = NN;  // padded tail -> gather row 0, scatter to dustbin row NN
    if (e < eend) {
      const int eo = order[e];
      s = nin[eo];
      o = nout[eo];
    }
    ldsSrc[threadIdx.x] = s;
    ldsOut[threadIdx.x] = o;
  }
  __syncthreads();

  // Stage 64 gathered X rows via async-to-LDS DMA (VGPR bypass).
  for (int idx = threadIdx.x; idx < 64 * 32; idx += 256) {
    const int rw = idx >> 5, c4 = idx & 31;
    async_b128_to_lds(ldsX + idx * 4, X + (size_t)ldsSrc[rw] * DD + c4 * 4);
  }
  // Stage W_r (cols r*128..r*128+127 of W_rel) pair-interleaved.
  stage_w_pairs(ldsW, Wrel + r * DD, RD);
  wait_async_all();
  __syncthreads();

  const int wave = threadIdx.x >> 5, lane = threadIdx.x & 31;
  const int row = lane & 15, hi = lane >> 4, n0 = wave * 16;

  v8f acc[4] = {};
#pragma unroll
  for (int kk = 0; kk < 32; ++kk) {
    const int kb = kk * 4 + 2 * hi;
    const v2f b = bfrag(ldsW, 2 * kk + hi, n0 + row);
#pragma unroll
    for (int t = 0; t < 4; ++t) {  // B-frag reused across 4 edge tiles
      const v2f a = *(const v2f*)(ldsX + (t * 16 + row) * DD + kb);  // ds_load_b64
      acc[t] = wmma4(a, b, acc[t]);
    }
  }
  const float bb = brel[r * DD + n0 + row];
#pragma unroll
  for (int t = 0; t < 4; ++t) {
#pragma unroll
    for (int i = 0; i < 8; ++i) {
      const int orow = ldsOut[t * 16 + i + 8 * hi];
      atomicAdd(&H[(size_t)orow * DD + n0 + row], acc[t][i] + bb);
    }
  }
}

__global__ __launch_bounds__(256) void k_relu_res(const float4* __restrict__ h,
                                                  const float4* __restrict__ x,
                                                  float4* __restrict__ o) {
  const size_t i = (size_t)blockIdx.x * 256 + threadIdx.x;
  const float4 hv = h[i], xv = x[i];
  o[i] = make_float4(fmaxf(hv.x, 0.f) + xv.x, fmaxf(hv.y, 0.f) + xv.y,
                     fmaxf(hv.z, 0.f) + xv.z, fmaxf(hv.w, 0.f) + xv.w);
}

__global__ __launch_bounds__(256) void k_zero_gf(float* gf) {
  for (int i = threadIdx.x; i < GG * DD; i += 256) gf[i] = 0.f;
}

// SumReadout: node2graph is sorted -> per-block running partial sums, few atomics.
__global__ __launch_bounds__(128) void k_readout(const float* __restrict__ nf,
                                                 const int* __restrict__ n2g,
                                                 float* __restrict__ gf) {
  __shared__ int sg[128];
  const int nbase = blockIdx.x * 128;
  sg[threadIdx.x] = n2g[nbase + threadIdx.x];
  __syncthreads();
  const int col = threadIdx.x;
  float sum = 0.f;
  int cur = sg[0];
  for (int i = 0; i < 128; ++i) {
    const int g = sg[i];
    if (g != cur) {
      atomicAdd(&gf[cur * DD + col], sum);
      sum = 0.f;
      cur = g;
    }
    sum += nf[(size_t)(nbase + i) * DD + col];
  }
  atomicAdd(&gf[cur * DD + col], sum);
}

extern "C" void kernel_launch(void* const* d_in, const int* in_sizes, int n_in,
                              void* d_out, int out_size, void* d_ws,
                              size_t ws_size, hipStream_t stream) {
  (void)in_sizes; (void)n_in; (void)out_size; (void)ws_size;
  const float* node_feat = (const float*)d_in[0];
  const int* nin = (const int*)d_in[1];
  const int* nout = (const int*)d_in[2];
  const int* rel = (const int*)d_in[3];
  const int* n2g = (const int*)d_in[4];
  const float* Wself[3] = {(const float*)d_in[5], (const float*)d_in[9],
                           (const float*)d_in[13]};
  const float* bself[3] = {(const float*)d_in[6], (const float*)d_in[10],
                           (const float*)d_in[14]};
  const float* Wrel[3] = {(const float*)d_in[7], (const float*)d_in[11],
                          (const float*)d_in[15]};
  const float* brel[3] = {(const float*)d_in[8], (const float*)d_in[12],
                          (const float*)d_in[16]};
  float* gf = (float*)d_out;            // [G, D]
  float* nf_out = gf + GG * DD;         // [N, D]

  char* ws = (char*)d_ws;
  const size_t sB = (size_t)(NN + 1) * DD * sizeof(float);  // +1 dustbin row
  float* B0 = (float*)ws;
  float* B1 = (float*)(ws + sB);
  int* order = (int*)(ws + 2 * sB);
  int* meta = (int*)(ws + 2 * sB + (size_t)NE * sizeof(int));

  k_meta_zero<<<1, 64, 0, stream>>>(meta);
  k_hist<<<NE / 256, 256, 0, stream>>>(rel, meta);
  k_scan<<<1, 64, 0, stream>>>(meta);
  k_fill<<<NE / 256, 256, 0, stream>>>(rel, meta, order);

  const int MSG_GRID = NE / 64 + RR;  // upper bound on total 64-edge tiles
  const size_t GEMM_LDS = (size_t)DD * DD * sizeof(float);              // 64 KB
  const size_t MSG_LDS = (size_t)(DD * DD + 64 * DD + 128) * sizeof(float);

  const float* xin = node_feat;
  for (int l = 0; l < 3; ++l) {
    float* h = (l == 1) ? B1 : B0;
    float* xout = (l == 0) ? B0 : (l == 1) ? B1 : nf_out;
    k_gemm_self<<<NN / 128, 256, GEMM_LDS, stream>>>(xin, Wself[l], bself[l], h);
    k_msg<<<MSG_GRID, 256, MSG_LDS, stream>>>(xin, Wrel[l], brel[l], order, nin,
                                              nout, meta, h);
    k_relu_res<<<(NN * DD / 4) / 256, 256, 0, stream>>>(
        (const float4*)h, (const float4*)xin, (float4*)xout);
    xin = xout;
  }
  k_zero_gf<<<1, 256, 0, stream>>>(gf);
  k_readout<<<NN / 128, 128, 0, stream>>>(nf_out, n2g, gf);
}